// GraphAttentionLayer_76132590289005
// MI455X (gfx1250) — compile-verified
//
#include <hip/hip_runtime.h>

// Problem constants (from reference): B=2, N=1024, Fin=64, F(out)=32
#define GAT_B   2
#define GAT_N   1024
#define GAT_FIN 64
#define GAT_F   32
#define ALPHA   0.01f

typedef __attribute__((ext_vector_type(2))) float v2f;
typedef __attribute__((ext_vector_type(8))) float v8f;

__device__ __forceinline__ float leaky(float x) {
    return (x >= 0.0f) ? x : ALPHA * x;
}

// ---------------------------------------------------------------------------
// Kernel 1: projections  h = inp@W, kk = k@Wk, vv = v@Wv   ([2048,64]@[64,32])
// One thread per output element; 3*2048*32 = 196608 threads.
// ---------------------------------------------------------------------------
__global__ void gat_proj_kernel(const float* __restrict__ inp,
                                const float* __restrict__ kin,
                                const float* __restrict__ vin,
                                const float* __restrict__ W,
                                const float* __restrict__ Wk,
                                const float* __restrict__ Wv,
                                float* __restrict__ h,
                                float* __restrict__ kk,
                                float* __restrict__ vv) {
    const int PER = GAT_B * GAT_N * GAT_F;  // 65536
    int idx = blockIdx.x * blockDim.x + threadIdx.x;
    int mat = idx / PER;          // 0 = h, 1 = kk, 2 = vv
    int rem = idx - mat * PER;
    int row = rem >> 5;           // b*N + n, 0..2047
    int col = rem & 31;
    const float* src = (mat == 0) ? inp : (mat == 1 ? kin : vin);
    const float* wm  = (mat == 0) ? W   : (mat == 1 ? Wk  : Wv);
    float*       dst = (mat == 0) ? h   : (mat == 1 ? kk  : vv);
    const float* srow = src + row * GAT_FIN;
    float acc = 0.0f;
#pragma unroll
    for (int c = 0; c < GAT_FIN; ++c)
        acc = fmaf(srow[c], wm[c * GAT_F + col], acc);
    dst[rem] = acc;
}

// ---------------------------------------------------------------------------
// Kernel 2: collapsed scores.
//  sH[b,r] = leaky( h[b,r,:] . (a[0:32]+a[32:64]) )           (rows i < 512)
//  sK[b,j] = leaky( kk[b,(2j)%N,:].a_lo + kk[b,(2j+1)%N,:].a_hi )  (i >= 512)
// 4096 threads total.
// ---------------------------------------------------------------------------
__global__ void gat_scores_kernel(const float* __restrict__ h,
                                  const float* __restrict__ kk,
                                  const float* __restrict__ a,
                                  float* __restrict__ sH,
                                  float* __restrict__ sK) {
    int idx = blockIdx.x * blockDim.x + threadIdx.x;  // 0..4095
    if (idx < GAT_B * GAT_N) {
        int b = idx >> 10, r = idx & (GAT_N - 1);
        const float* hr = h + (b * GAT_N + r) * GAT_F;
        float acc = 0.0f;
#pragma unroll
        for (int c = 0; c < GAT_F; ++c)
            acc = fmaf(hr[c], a[c] + a[c + GAT_F], acc);
        sH[idx] = leaky(acc);
    } else {
        int t = idx - GAT_B * GAT_N;
        int b = t >> 10, j = t & (GAT_N - 1);
        int j0 = (2 * j) & (GAT_N - 1);
        int j1 = (2 * j + 1) & (GAT_N - 1);
        const float* k0r = kk + (b * GAT_N + j0) * GAT_F;
        const float* k1r = kk + (b * GAT_N + j1) * GAT_F;
        float acc = 0.0f;
#pragma unroll
        for (int c = 0; c < GAT_F; ++c) {
            acc = fmaf(k0r[c], a[c], acc);
            acc = fmaf(k1r[c], a[c + GAT_F], acc);
        }
        sK[t] = leaky(acc);
    }
}

// ---------------------------------------------------------------------------
// Kernel 3: attention prep. 2 blocks (one per batch) x 1024 threads.
//  - rows i<512: two-value softmax in closed form -> attLo/attHi
//  - rows i>=512: one shared softmax of sK -> softK
// ---------------------------------------------------------------------------
__global__ void gat_att_kernel(const float* __restrict__ sH,
                               const float* __restrict__ sK,
                               float* __restrict__ attLo,
                               float* __restrict__ attHi,
                               float* __restrict__ softK) {
    __shared__ float red[GAT_N];
    int b = blockIdx.x;
    int tid = threadIdx.x;

    if (tid < GAT_N / 2) {
        float v0 = sH[b * GAT_N + 2 * tid];
        float v1 = sH[b * GAT_N + 2 * tid + 1];
        float m  = fmaxf(v0, v1);
        float e0 = expf(v0 - m);
        float e1 = expf(v1 - m);
        float d  = (float)(GAT_N / 2) * (e0 + e1);
        attLo[b * (GAT_N / 2) + tid] = e0 / d;
        attHi[b * (GAT_N / 2) + tid] = e1 / d;
    }

    float x = sK[b * GAT_N + tid];
    red[tid] = x;
    __syncthreads();
    for (int s = GAT_N / 2; s > 0; s >>= 1) {
        if (tid < s) red[tid] = fmaxf(red[tid], red[tid + s]);
        __syncthreads();
    }
    float mx = red[0];
    __syncthreads();
    float e = expf(x - mx);
    red[tid] = e;
    __syncthreads();
    for (int s = GAT_N / 2; s > 0; s >>= 1) {
        if (tid < s) red[tid] += red[tid + s];
        __syncthreads();
    }
    softK[b * GAT_N + tid] = e / red[0];
}

// ---------------------------------------------------------------------------
// Kernel 4: out = leaky( ((att + adj) * 0.5) @ vv ) via V_WMMA_F32_16X16X4_F32.
// 256 waves total: b(2) x Mtile(64) x Ntile(2). K-loop = 256 WMMAs/wave.
// A-fragment (16x4 f32, 2 VGPRs): lane l -> M = i0+(l&15), K = k0+2*(l>>4)+{0,1}.
// att is fused into the A fragment build (never materialized).
// ---------------------------------------------------------------------------
__global__ __launch_bounds__(256) void gat_wmma_kernel(
        const float* __restrict__ adj,
        const float* __restrict__ vv,
        const float* __restrict__ attLo,
        const float* __restrict__ attHi,
        const float* __restrict__ softK,
        float* __restrict__ out) {
    int wid  = (blockIdx.x * blockDim.x + threadIdx.x) >> 5;  // 0..255
    int lane = threadIdx.x & 31;
    int b    = wid >> 7;
    int rem  = wid & 127;
    int i0   = (rem >> 1) * 16;   // M tile base
    int n0   = (rem & 1) * 16;    // N tile base

    int mrow  = i0 + (lane & 15);
    int khalf = lane >> 4;        // 0 or 1
    int ncol  = n0 + (lane & 15);

    const float* vvb = vv + b * (GAT_N * GAT_F);
    const float* skb = softK + b * GAT_N;
    const float* adjRow = adj + mrow * GAT_N;

    bool topHalf = (mrow < GAT_N / 2);
    int  li   = topHalf ? mrow : 0;
    float aLoR = attLo[b * (GAT_N / 2) + li];
    float aHiR = attHi[b * (GAT_N / 2) + li];

    v8f acc = {};
    for (int k0 = 0; k0 < GAT_N; k0 += 4) {
        int kb = k0 + khalf * 2;
        // stream adj ahead (global_prefetch_b8)
        if ((k0 & 63) == 0)
            __builtin_prefetch(adjRow + k0 + 64, 0, 1);

        float2 ad = *(const float2*)(adjRow + kb);   // 8B aligned (kb even)
        float2 sk = *(const float2*)(skb + kb);
        float av   = (kb < GAT_N / 2) ? aLoR : aHiR;
        float attx = topHalf ? av : sk.x;            // branchless cndmask
        float atty = topHalf ? av : sk.y;

        v2f afrag;
        afrag.x = (attx + ad.x) * 0.5f;
        afrag.y = (atty + ad.y) * 0.5f;

        v2f bfrag;                                   // B 4x16: K=kb,kb+1 col=ncol
        bfrag.x = vvb[kb * GAT_F + ncol];
        bfrag.y = vvb[(kb + 1) * GAT_F + ncol];

        acc = __builtin_amdgcn_wmma_f32_16x16x4_f32(
                  false, afrag, false, bfrag, (short)0, acc, false, false);
    }

    // C/D layout: VGPR r -> row i0 + r + 8*khalf, col = n0 + (lane&15)
    float* ob = out + b * (GAT_N * GAT_F);
#pragma unroll
    for (int r = 0; r < 8; ++r) {
        int row = i0 + r + khalf * 8;
        float vl = acc[r];
        vl = (vl >= 0.0f) ? vl : ALPHA * vl;
        ob[row * GAT_F + ncol] = vl;
    }
}

// ---------------------------------------------------------------------------
extern "C" void kernel_launch(void* const* d_in, const int* in_sizes, int n_in,
                              void* d_out, int out_size, void* d_ws, size_t ws_size,
                              hipStream_t stream) {
    const float* inp = (const float*)d_in[0];
    const float* kin = (const float*)d_in[1];
    const float* vin = (const float*)d_in[2];
    const float* adj = (const float*)d_in[3];
    const float* W   = (const float*)d_in[4];
    const float* Wk  = (const float*)d_in[5];
    const float* Wv  = (const float*)d_in[6];
    const float* a   = (const float*)d_in[7];
    float* out = (float*)d_out;

    float* ws    = (float*)d_ws;
    float* h     = ws;                       // 65536
    float* kk    = h  + GAT_B * GAT_N * GAT_F;   // 65536
    float* vv    = kk + GAT_B * GAT_N * GAT_F;   // 65536
    float* sH    = vv + GAT_B * GAT_N * GAT_F;   // 2048
    float* sK    = sH + GAT_B * GAT_N;           // 2048
    float* attLo = sK + GAT_B * GAT_N;           // 1024
    float* attHi = attLo + GAT_B * (GAT_N / 2);  // 1024
    float* softK = attHi + GAT_B * (GAT_N / 2);  // 2048

    gat_proj_kernel<<<768, 256, 0, stream>>>(inp, kin, vin, W, Wk, Wv, h, kk, vv);
    gat_scores_kernel<<<16, 256, 0, stream>>>(h, kk, a, sH, sK);
    gat_att_kernel<<<GAT_B, GAT_N, 0, stream>>>(sH, sK, attLo, attHi, softK);
    gat_wmma_kernel<<<32, 256, 0, stream>>>(adj, vv, attLo, attHi, softK, out);
}